// GraphEncoder_65463891526244
// MI455X (gfx1250) — compile-verified
//
#include <hip/hip_runtime.h>

typedef float v2f __attribute__((ext_vector_type(2)));
typedef float v8f __attribute__((ext_vector_type(8)));

#define N_NODES 100000
#define N_EDGES 640000
#define DIM     128
#define N_LAYERS 3
#define N_MSG   (N_EDGES + N_NODES)   // edges + self-loops

// ---------------------------------------------------------------------------
// Degree / normalization precompute (layer-invariant)
// ---------------------------------------------------------------------------
__global__ void zero_f32(float* p, int n) {
    int i = blockIdx.x * blockDim.x + threadIdx.x;
    if (i < n) p[i] = 0.0f;
}

__global__ void deg_count(const int* __restrict__ e_dst, float* __restrict__ deg) {
    int i = blockIdx.x * blockDim.x + threadIdx.x;
    if (i >= N_MSG) return;
    int d = (i < N_EDGES) ? e_dst[i] : (i - N_EDGES);   // self-loop tail
    atomicAdd(&deg[d], 1.0f);
}

__global__ void dinv_kernel(const float* __restrict__ deg, float* __restrict__ dinv) {
    int i = blockIdx.x * blockDim.x + threadIdx.x;
    if (i >= N_NODES) return;
    float d = deg[i];
    dinv[i] = (d > 0.0f) ? rsqrtf(d) : 0.0f;
}

__global__ void norm_kernel(const int* __restrict__ e_src, const int* __restrict__ e_dst,
                            const float* __restrict__ dinv, float* __restrict__ norm) {
    int i = blockIdx.x * blockDim.x + threadIdx.x;
    if (i >= N_MSG) return;
    int s, d;
    if (i < N_EDGES) { s = e_src[i]; d = e_dst[i]; }
    else             { s = d = i - N_EDGES; }
    norm[i] = dinv[s] * dinv[d];
}

// ---------------------------------------------------------------------------
// Dense feature transform: Out[N,128] = A[N,128] @ W[128,128], full f32 via
// V_WMMA_F32_16X16X4_F32.  One wave per 16x16 output tile; 8 waves per block
// cover one 16-row stripe across all 128 output columns.  K=128 -> 32 WMMAs.
// ---------------------------------------------------------------------------
__global__ __launch_bounds__(256) void gemm_wmma_f32(
    const float* __restrict__ A, const float* __restrict__ W,
    float* __restrict__ Out, int n_rows)
{
    const int lane   = threadIdx.x & 31;
    const int wave   = threadIdx.x >> 5;          // col tile 0..7
    const int row0   = blockIdx.x * 16;           // row tile base
    if (row0 >= n_rows) return;                   // block-uniform: EXEC stays full

    const int lrow   = lane & 15;
    const int hiHalf = lane >> 4;                 // 0: lanes 0-15, 1: lanes 16-31
    const int m      = row0 + lrow;               // A row for this lane
    const int col    = wave * 16 + lrow;          // B/D column for this lane

    const float* arow = A + (size_t)m * DIM;

    v8f c = {0.f, 0.f, 0.f, 0.f, 0.f, 0.f, 0.f, 0.f};

    #pragma unroll 4
    for (int k0 = 0; k0 < DIM; k0 += 4) {
        const int ka = k0 + hiHalf * 2;           // this lane's K pair (even -> 8B aligned)
        v2f a = *(const v2f*)(arow + ka);         // A[m][ka], A[m][ka+1]
        v2f b;
        b.x = W[(size_t)ka       * DIM + col];    // B row ka
        b.y = W[(size_t)(ka + 1) * DIM + col];    // B row ka+1
        // (neg_a, A, neg_b, B, c_mod, C, reuse_a, reuse_b)
        c = __builtin_amdgcn_wmma_f32_16x16x4_f32(false, a, false, b,
                                                  (short)0, c, false, false);
    }

    // C/D layout: VGPR r -> row (r + 8*hiHalf), N = lane&15
    #pragma unroll
    for (int r = 0; r < 8; ++r) {
        const int rowM = row0 + r + hiHalf * 8;
        Out[(size_t)rowM * DIM + col] = c[r];
    }
}

// ---------------------------------------------------------------------------
// acc[n][d] = bias[d]  (bias added before scatter-sum, matching reference)
// ---------------------------------------------------------------------------
__global__ void acc_init(float* __restrict__ acc, const float* __restrict__ bias) {
    size_t i = (size_t)blockIdx.x * blockDim.x + threadIdx.x;
    if (i >= (size_t)N_NODES * DIM) return;
    acc[i] = bias[i & (DIM - 1)];
}

// ---------------------------------------------------------------------------
// Scatter: one wave per message; lane handles 4 contiguous floats (32*4=128).
// acc[dst] += norm * hw[src]  via global_atomic_add_f32.
// ---------------------------------------------------------------------------
__global__ __launch_bounds__(256) void scatter_edges(
    const float* __restrict__ hw, const float* __restrict__ norm,
    const int* __restrict__ e_src, const int* __restrict__ e_dst,
    float* __restrict__ acc)
{
    const int lane = threadIdx.x & 31;
    const int eid  = blockIdx.x * (blockDim.x >> 5) + (threadIdx.x >> 5);
    if (eid >= N_MSG) return;

    int s, d;
    if (eid < N_EDGES) { s = e_src[eid]; d = e_dst[eid]; }
    else               { s = d = eid - N_EDGES; }

    const float w = norm[eid];
    const float4 v = *(const float4*)(hw + (size_t)s * DIM + lane * 4);
    float* dp = acc + (size_t)d * DIM + lane * 4;
    atomicAdd(dp + 0, w * v.x);
    atomicAdd(dp + 1, w * v.y);
    atomicAdd(dp + 2, w * v.z);
    atomicAdd(dp + 3, w * v.w);
}

__global__ void relu_inplace(float* __restrict__ p, size_t n) {
    size_t i = (size_t)blockIdx.x * blockDim.x + threadIdx.x;
    if (i < n) p[i] = fmaxf(p[i], 0.0f);
}

__global__ void relu_out(const float* __restrict__ in, float* __restrict__ out, size_t n) {
    size_t i = (size_t)blockIdx.x * blockDim.x + threadIdx.x;
    if (i < n) out[i] = fmaxf(in[i], 0.0f);
}

// ---------------------------------------------------------------------------
// kernel_launch
// ---------------------------------------------------------------------------
extern "C" void kernel_launch(void* const* d_in, const int* in_sizes, int n_in,
                              void* d_out, int out_size, void* d_ws, size_t ws_size,
                              hipStream_t stream)
{
    const float* x    = (const float*)d_in[0];   // [1, N, 128]
    const int*   edge = (const int*)d_in[1];     // [2, E]
    const float* Ws   = (const float*)d_in[2];   // [3, 128, 128]
    const float* bs   = (const float*)d_in[3];   // [3, 128]
    float*       out  = (float*)d_out;

    const int* e_src = edge;
    const int* e_dst = edge + N_EDGES;

    // Workspace layout (floats)
    float* P0   = (float*)d_ws;                      // acc / h ping buffer  [N*DIM]
    float* P1   = P0 + (size_t)N_NODES * DIM;        // hw buffer            [N*DIM]
    float* deg  = P1 + (size_t)N_NODES * DIM;        // [N]
    float* dinv = deg + N_NODES;                     // [N]
    float* norm = dinv + N_NODES;                    // [N_MSG]

    const size_t NELEM = (size_t)N_NODES * DIM;
    const int B = 256;

    // --- layer-invariant normalization ---
    zero_f32<<<(N_NODES + B - 1) / B, B, 0, stream>>>(deg, N_NODES);
    deg_count<<<(N_MSG + B - 1) / B, B, 0, stream>>>(e_dst, deg);
    dinv_kernel<<<(N_NODES + B - 1) / B, B, 0, stream>>>(deg, dinv);
    norm_kernel<<<(N_MSG + B - 1) / B, B, 0, stream>>>(e_src, e_dst, dinv, norm);

    // --- 3 GCN layers ---
    const float* h_in = x;
    for (int l = 0; l < N_LAYERS; ++l) {
        // hw = h_in @ W_l   (WMMA f32)
        gemm_wmma_f32<<<N_NODES / 16, 256, 0, stream>>>(
            h_in, Ws + (size_t)l * DIM * DIM, P1, N_NODES);

        // acc = bias
        acc_init<<<(unsigned)((NELEM + B - 1) / B), B, 0, stream>>>(P0, bs + (size_t)l * DIM);

        // acc[dst] += norm * hw[src]
        scatter_edges<<<(N_MSG + 7) / 8, 256, 0, stream>>>(P1, norm, e_src, e_dst, P0);

        if (l < N_LAYERS - 1) {
            relu_inplace<<<(unsigned)((NELEM + B - 1) / B), B, 0, stream>>>(P0, NELEM);
            h_in = P0;
        } else {
            relu_out<<<(unsigned)((NELEM + B - 1) / B), B, 0, stream>>>(P0, out, NELEM);
        }
    }
}